// Dependence_Learning_44160853737976
// MI455X (gfx1250) — compile-verified
//
#include <hip/hip_runtime.h>

typedef __attribute__((ext_vector_type(2))) float v2f;
typedef __attribute__((ext_vector_type(4))) float v4f;
typedef __attribute__((ext_vector_type(8))) float v8f;

#define CD 256      // channels (in = out)
#define LD 4096     // H*W
#define BB 4        // batch
#define NY (BB*CD*LD)
#define BPAD 18     // padded B-panel row (bank-conflict-free, keeps 8B align)

// ---------------------------------------------------------------------------
// Async global->LDS staging (CDNA5): per-lane LDS dest offset, per-lane 64-bit
// global address (GV mode). Tracked by ASYNCcnt.
// ---------------------------------------------------------------------------
__device__ __forceinline__ void async_b128(unsigned ldsOff, const void* gaddr) {
    asm volatile("global_load_async_to_lds_b128 %0, %1, off"
                 :: "v"(ldsOff), "v"(gaddr) : "memory");
}
__device__ __forceinline__ void async_b32(unsigned ldsOff, const void* gaddr) {
    asm volatile("global_load_async_to_lds_b32 %0, %1, off"
                 :: "v"(ldsOff), "v"(gaddr) : "memory");
}

// ---------------------------------------------------------------------------
// GEMM: Y[b,o,l] = sum_c W[o,c] * X[b,c,l] + bias[o]
// fp32 WMMA 16x16x4. 256 threads = 8 waves; waves 4(M) x 2(N); each wave owns
// a 16(M) x 32(N) strip (two v8f accumulators). WG tile 64x64. K panels of 16
// double-buffered in LDS via async copies; B panel stored transposed [n][k]
// (pad 18) so A and B fragments are single contiguous ds_load_b64 reads.
// ---------------------------------------------------------------------------
__global__ __launch_bounds__(256) void gemm_bias(
    const float* __restrict__ W, const float* __restrict__ X,
    const float* __restrict__ bias, float* __restrict__ Y)
{
    __shared__ __align__(16) float As[2][64 * 16];     // [row][k]
    __shared__ __align__(16) float Bs[2][64 * BPAD];   // [n][k] transposed

    const int b       = blockIdx.z;
    const int rowBase = blockIdx.y * 64;
    const int colBase = blockIdx.x * 64;
    const int tid     = threadIdx.x;
    const int lane    = tid & 31;         // wave32
    const int wave    = tid >> 5;
    const int waveM   = wave >> 1;        // 0..3
    const int waveN   = wave & 1;         // 0..1
    const int m       = lane & 15;
    const int half    = lane >> 4;

    const float* Xb = X + (size_t)b * CD * LD;
    float*       Yb = Y + (size_t)b * CD * LD;

    v8f acc0 = {};
    v8f acc1 = {};

    // staging indices
    const int ar  = tid >> 2;             // 0..63  A row
    const int ac  = (tid & 3) << 2;       // 0,4,8,12
    const int bkr = tid >> 4;             // 0..15  B k-row
    const int bcc = (tid & 15) << 2;      // 0..60  B col base

    // stage panel kt into buffer buf: 1 x b128 (A) + 4 x b32 transpose (B)
    auto stage = [&](int kt, int buf) {
        async_b128((unsigned)(size_t)(&As[buf][ar * 16 + ac]),
                   (const void*)(W + (size_t)(rowBase + ar) * CD + kt + ac));
        #pragma unroll
        for (int j = 0; j < 4; ++j) {
            const int n = bcc + j;
            async_b32((unsigned)(size_t)(&Bs[buf][n * BPAD + bkr]),
                      (const void*)(Xb + (size_t)(kt + bkr) * LD + colBase + n));
        }
    };

    stage(0, 0);
    for (int it = 0; it < 16; ++it) {
        const int buf = it & 1;
        if (it + 1 < 16) {
            stage((it + 1) * 16, buf ^ 1);
            // in-order completion: <=5 outstanding => previous stage landed
            asm volatile("s_wait_asynccnt 0x5" ::: "memory");
        } else {
            asm volatile("s_wait_asynccnt 0x0" ::: "memory");
        }
        __syncthreads();

        const float* ap  = &As[buf][(waveM * 16 + m) * 16];
        const float* bp0 = &Bs[buf][(waveN * 32 + m) * BPAD];
        const float* bp1 = bp0 + 16 * BPAD;
        #pragma unroll
        for (int k0 = 0; k0 < 16; k0 += 4) {
            const int kb = k0 + 2 * half;           // K pair {kb, kb+1}
            v2f a  = *(const v2f*)(ap  + kb);
            v2f b0 = *(const v2f*)(bp0 + kb);
            v2f b1 = *(const v2f*)(bp1 + kb);
            acc0 = __builtin_amdgcn_wmma_f32_16x16x4_f32(
                false, a, false, b0, (short)0, acc0, false, false);
            acc1 = __builtin_amdgcn_wmma_f32_16x16x4_f32(
                false, a, false, b1, (short)0, acc1, false, false);
        }
        __syncthreads();   // buffer `buf` safe to overwrite by stage(it+2)
    }

    // C/D layout: VGPR v -> row (v + 8*half), col n = lane&15
    union { v8f v; float f[8]; } u0, u1;
    u0.v = acc0; u1.v = acc1;
    const int row0 = rowBase + waveM * 16;
    const int col0 = colBase + waveN * 32;
    #pragma unroll
    for (int v = 0; v < 8; ++v) {
        const int row = row0 + v + half * 8;
        const float bi = bias[row];
        Yb[(size_t)row * LD + col0 + m]      = u0.f[v] + bi;
        Yb[(size_t)row * LD + col0 + 16 + m] = u1.f[v] + bi;
    }
}

// ---------------------------------------------------------------------------
// Training-mode BN stats per channel over (B,H,W): one WG per channel.
// ---------------------------------------------------------------------------
__global__ __launch_bounds__(256) void bn_stats(
    const float* __restrict__ Y, const float* __restrict__ gamma,
    const float* __restrict__ beta, float* __restrict__ scale,
    float* __restrict__ shift)
{
    __shared__ float ssum[256];
    __shared__ float ssq[256];
    const int o = blockIdx.x;
    const int tid = threadIdx.x;
    float s = 0.f, s2 = 0.f;
    for (int idx = tid; idx < BB * LD; idx += 256) {
        const int b = idx >> 12;
        const int l = idx & (LD - 1);
        const float v = Y[((size_t)b * CD + o) * LD + l];
        s += v; s2 += v * v;
    }
    ssum[tid] = s; ssq[tid] = s2;
    __syncthreads();
    for (int off = 128; off > 0; off >>= 1) {
        if (tid < off) { ssum[tid] += ssum[tid + off]; ssq[tid] += ssq[tid + off]; }
        __syncthreads();
    }
    if (tid == 0) {
        const float inv = 1.f / (float)(BB * LD);
        const float mu  = ssum[0] * inv;
        const float var = ssq[0] * inv - mu * mu;
        const float sc  = gamma[o] * rsqrtf(var + 1e-5f);
        scale[o] = sc;
        shift[o] = beta[o] - mu * sc;
    }
}

__global__ __launch_bounds__(256) void bn_apply(
    const float* __restrict__ Y, const float* __restrict__ scale,
    const float* __restrict__ shift, float* __restrict__ T)
{
    const size_t i = (size_t)blockIdx.x * 256 + threadIdx.x;
    const int o = (int)((i >> 12) & (CD - 1));
    const float v = Y[i] * scale[o] + shift[o];
    T[i] = v > 0.f ? v : 0.f;
}

// ---------------------------------------------------------------------------
// Attention stage: per (b,c), flat space g in [0, 9L):
//   F(t,g) = 3x3-unfold of tile t at tap k=g/L, pos l=g%L (zero-padded)
//   logit[g] = F1[g]*F2[gc] + F1[gc]*F2[g],  gc = 9*(g/9)+4
//   w = softmax over each contiguous 9-group
//   pre[l] = sum_k w[k*L+l] * Fx[k*L+l]
// One WG (256 thr) per (b,c); tiles + group stats in LDS (80 KB).
// ---------------------------------------------------------------------------
__device__ __forceinline__ float unf3x3(const float* t, int g) {
    const int k  = g >> 12;            // tap 0..8  (L = 4096)
    const int l  = g & (LD - 1);
    const int hi = (l >> 6) + (k / 3) - 1;
    const int wi = (l & 63) + (k % 3) - 1;
    if ((unsigned)hi < 64u && (unsigned)wi < 64u) return t[(hi << 6) + wi];
    return 0.f;
}

__global__ __launch_bounds__(256) void attn(
    const float* __restrict__ X, const float* __restrict__ T1,
    const float* __restrict__ T2, float* __restrict__ PRE)
{
    __shared__ float sx[LD];
    __shared__ float s1[LD];
    __shared__ float s2[LD];
    __shared__ float gmax[LD];
    __shared__ float ginv[LD];

    const int c = blockIdx.x;
    const int b = blockIdx.y;
    const int tid = threadIdx.x;
    const size_t base = ((size_t)b * CD + c) * LD;

    for (int i = tid; i < LD; i += 256) {
        sx[i] = X[base + i];
        s1[i] = T1[base + i];
        s2[i] = T2[base + i];
    }
    __syncthreads();

    // pass 1: per-group max and 1/sum of exp
    for (int q = tid; q < LD; q += 256) {
        const int gc = 9 * q + 4;
        const float f1c = unf3x3(s1, gc);
        const float f2c = unf3x3(s2, gc);
        float lg[9];
        float mx = -3.4e38f;
        #pragma unroll
        for (int j = 0; j < 9; ++j) {
            const int g = 9 * q + j;
            const float lo = unf3x3(s1, g) * f2c + f1c * unf3x3(s2, g);
            lg[j] = lo;
            mx = lo > mx ? lo : mx;
        }
        float sum = 0.f;
        #pragma unroll
        for (int j = 0; j < 9; ++j) sum += __expf(lg[j] - mx);
        gmax[q] = mx;
        ginv[q] = 1.f / sum;
    }
    __syncthreads();

    // pass 2: weighted 3x3 gather of x
    for (int l = tid; l < LD; l += 256) {
        float acc = 0.f;
        #pragma unroll
        for (int k = 0; k < 9; ++k) {
            const int g  = (k << 12) + l;
            const int q  = g / 9;
            const int gc = 9 * q + 4;
            const float lo = unf3x3(s1, g) * unf3x3(s2, gc)
                           + unf3x3(s1, gc) * unf3x3(s2, g);
            const float w = __expf(lo - gmax[q]) * ginv[q];
            acc += w * unf3x3(sx, g);
        }
        PRE[base + l] = acc;
    }
}

// ---------------------------------------------------------------------------
extern "C" void kernel_launch(void* const* d_in, const int* in_sizes, int n_in,
                              void* d_out, int out_size, void* d_ws, size_t ws_size,
                              hipStream_t stream) {
    const float* x   = (const float*)d_in[0];
    const float* w1  = (const float*)d_in[1];
    const float* b1  = (const float*)d_in[2];
    const float* g1  = (const float*)d_in[3];
    const float* be1 = (const float*)d_in[4];
    const float* w2  = (const float*)d_in[5];
    const float* b2  = (const float*)d_in[6];
    const float* g2  = (const float*)d_in[7];
    const float* be2 = (const float*)d_in[8];
    const float* wf  = (const float*)d_in[9];
    const float* bf  = (const float*)d_in[10];
    const float* gf  = (const float*)d_in[11];
    const float* bef = (const float*)d_in[12];

    float* ws    = (float*)d_ws;
    float* Y     = ws;                 // [B,C,L]
    float* T1    = ws + (size_t)NY;
    float* T2    = ws + (size_t)2 * NY;
    float* PRE   = ws + (size_t)3 * NY;
    float* scale = ws + (size_t)4 * NY;
    float* shift = scale + CD;

    const dim3 gemmGrid(LD / 64, CD / 64, BB);   // (64, 4, 4)
    const dim3 attnGrid(CD, BB);                 // (256, 4)
    const int  ewBlocks = NY / 256;              // 16384

    // tri1 = relu(bn(conv1x1(x, w1)))
    gemm_bias<<<gemmGrid, 256, 0, stream>>>(w1, x, b1, Y);
    bn_stats <<<CD, 256, 0, stream>>>(Y, g1, be1, scale, shift);
    bn_apply <<<ewBlocks, 256, 0, stream>>>(Y, scale, shift, T1);

    // tri2 = relu(bn(conv1x1(x, w2)))
    gemm_bias<<<gemmGrid, 256, 0, stream>>>(w2, x, b2, Y);
    bn_stats <<<CD, 256, 0, stream>>>(Y, g2, be2, scale, shift);
    bn_apply <<<ewBlocks, 256, 0, stream>>>(Y, scale, shift, T2);

    // pre = softmax-weighted 3x3 gather
    attn<<<attnGrid, 256, 0, stream>>>(x, T1, T2, PRE);

    // out = relu(bn(conv1x1(pre, wf)))
    gemm_bias<<<gemmGrid, 256, 0, stream>>>(wf, PRE, bf, Y);
    bn_stats <<<CD, 256, 0, stream>>>(Y, gf, bef, scale, shift);
    bn_apply <<<ewBlocks, 256, 0, stream>>>(Y, scale, shift, (float*)d_out);
}